// EnhancedGraphEncoder_71992241815698
// MI455X (gfx1250) — compile-verified
//
#include <hip/hip_runtime.h>
#include <hip/hip_bf16.h>

typedef float v2f __attribute__((ext_vector_type(2)));
typedef float v4f __attribute__((ext_vector_type(4)));
typedef float v8f __attribute__((ext_vector_type(8)));

#define NNODES 50000
#define NEDGES 800000
#define NGRAPH 64
#define DIN    128
#define DH     256
#define DOUT   128

#define KC    128   // K chunk staged in LDS per barrier phase
#define PITCH 80    // LDS pitch in float2 units per K-pair row:
                    // 2*PITCH = 160 dwords == 32 mod 64 banks -> the two lane
                    // halves (pair p vs p+1) hit disjoint 32-bank halves.

// ---------------------------------------------------------------------------
// fp32 WMMA GEMM:  H[nrows x ncols] = X[nrows x K] @ W[K x ncols]
// block = 8 waves, each wave = 16 rows x 64 cols (4 accumulators)
// -> block tile = 128 rows x 64 cols, W tile staged once in LDS (8x reuse).
// LDS layout: wlds[p*PITCH + c] = { W[kbase+2p][colb+c], W[kbase+2p+1][colb+c] }
// so one ds_load_b64 delivers a lane's exact B fragment (its two K values).
// ---------------------------------------------------------------------------
__global__ __launch_bounds__(256) void gemm_wmma_f32(
    const float* __restrict__ X, const float* __restrict__ W,
    float* __restrict__ H, int nrows, int K, int ncols)
{
    __shared__ v2f wlds[(KC / 2) * PITCH];          // 64*80*8B = 40 KB

    const int lane = threadIdx.x & 31;
    const int wave = threadIdx.x >> 5;
    const int row0 = (blockIdx.x * 8 + wave) * 16;  // wave-uniform
    const int colb = blockIdx.y * 64;
    const int m    = lane & 15;                     // A row / B col within tile
    const int hk   = lane >> 4;                     // lane half: K pair select
    const bool active = (row0 < nrows);

    v8f acc0 = {}, acc1 = {}, acc2 = {}, acc3 = {};
    const float* xrow = X + (size_t)(row0 + m) * K;

    for (int kbase = 0; kbase < K; kbase += KC) {
        // ---- cooperative stage of W chunk into LDS (all 256 threads) ----
        for (int idx = threadIdx.x; idx < (KC / 2) * 32; idx += 256) {
            int p  = idx >> 5;                       // K pair within chunk
            int c2 = (idx & 31) << 1;                // starting col (even)
            const float* wr0 = W + (size_t)(kbase + 2 * p) * ncols + colb + c2;
            v2f g0 = *(const v2f*)wr0;               // W[2p][c2..c2+1]
            v2f g1 = *(const v2f*)(wr0 + ncols);     // W[2p+1][c2..c2+1]
            v4f st; st.x = g0.x; st.y = g1.x; st.z = g0.y; st.w = g1.y;
            *(v4f*)&wlds[p * PITCH + c2] = st;       // 16B aligned: c2 even, PITCH even
        }
        __syncthreads();

        if (active) {                                // wave-uniform: EXEC all-1s
#pragma unroll 4
            for (int k = 0; k < KC; k += 4) {
                // A fragment: one b64 load, lane m holds X[row, k+2hk .. k+2hk+1]
                v2f a = *(const v2f*)(xrow + kbase + k + 2 * hk);
                // B fragments: one ds_load_b64 each, conflict-free
                const v2f* wp = &wlds[((k >> 1) + hk) * PITCH + m];
                v2f b0 = wp[0], b1 = wp[16], b2 = wp[32], b3 = wp[48];
                acc0 = __builtin_amdgcn_wmma_f32_16x16x4_f32(false, a, false, b0, (short)0, acc0, false, false);
                acc1 = __builtin_amdgcn_wmma_f32_16x16x4_f32(false, a, false, b1, (short)0, acc1, false, false);
                acc2 = __builtin_amdgcn_wmma_f32_16x16x4_f32(false, a, false, b2, (short)0, acc2, false, false);
                acc3 = __builtin_amdgcn_wmma_f32_16x16x4_f32(false, a, false, b3, (short)0, acc3, false, false);
            }
        }
        __syncthreads();
    }

    if (active) {
        // C/D layout: VGPR v, lanes 0-15 -> M=v, lanes 16-31 -> M=v+8, N=lane&15
        const int mhi = hk * 8;
        for (int v = 0; v < 8; ++v) {
            float* hrow = H + (size_t)(row0 + mhi + v) * ncols + colb + m;
            hrow[0]  = acc0[v];
            hrow[16] = acc1[v];
            hrow[32] = acc2[v];
            hrow[48] = acc3[v];
        }
    }
}

// ---------------------------------------------------------------------------
// utility kernels
// ---------------------------------------------------------------------------
__global__ void k_zero(float* __restrict__ p, long long n) {
    long long i = (long long)blockIdx.x * blockDim.x + threadIdx.x;
    if (i < n) p[i] = 0.0f;
}

__global__ void k_count_deg(const int* __restrict__ ei, float* __restrict__ deg, int ne) {
    int e = blockIdx.x * blockDim.x + threadIdx.x;
    if (e < ne) atomicAdd(&deg[ei[ne + e]], 1.0f);   // dst row
}

__global__ void k_deg_to_dinv(float* __restrict__ deg, int n) {
    int i = blockIdx.x * blockDim.x + threadIdx.x;
    if (i < n) deg[i] = rsqrtf(deg[i] + 1.0f);       // +1 self loop
}

__global__ void k_edge_coef(const int* __restrict__ ei, const float* __restrict__ dinv,
                            float* __restrict__ coef, int ne) {
    int e = blockIdx.x * blockDim.x + threadIdx.x;
    if (e < ne) coef[e] = dinv[ei[e]] * dinv[ei[ne + e]];
}

// O += H * self_coef + b + Rb   (O already holds X @ Rw)
__global__ void k_combine(float* __restrict__ O, const float* __restrict__ H,
                          const float* __restrict__ dinv,
                          const float* __restrict__ b, const float* __restrict__ Rb,
                          int nrows, int d)
{
    long long idx = (long long)blockIdx.x * blockDim.x + threadIdx.x;
    if (idx >= (long long)nrows * d) return;
    int row = (int)(idx / d);
    int col = (int)(idx % d);
    float di = dinv[row];
    O[idx] += H[idx] * (di * di) + b[col] + Rb[col];
}

// O[dst] += H[src] * edge_coef : 1 thread per (edge, 4 features)
// b128 gather of H[src], coalesced per-float atomics on O[dst]
__global__ void k_scatter(const float* __restrict__ H, const int* __restrict__ ei,
                          const float* __restrict__ coef, float* __restrict__ O,
                          int ne, int d)
{
    const int dq = d >> 2;
    long long idx = (long long)blockIdx.x * blockDim.x + threadIdx.x;
    if (idx >= (long long)ne * dq) return;
    int e  = (int)(idx / dq);
    int f4 = (int)(idx % dq) << 2;
    int s = ei[e];
    int t = ei[ne + e];
    float c = coef[e];
    v4f h = *(const v4f*)&H[(size_t)s * d + f4];
    float* o = &O[(size_t)t * d + f4];
    atomicAdd(o + 0, h.x * c);
    atomicAdd(o + 1, h.y * c);
    atomicAdd(o + 2, h.z * c);
    atomicAdd(o + 3, h.w * c);
}

// LayerNorm (+ optional SiLU), one wave32 per row, shuffle reductions
__global__ __launch_bounds__(256) void k_layernorm(
    float* __restrict__ Hio, const float* __restrict__ g, const float* __restrict__ be,
    int nrows, int d, int act)
{
    int row = blockIdx.x * 8 + (threadIdx.x >> 5);
    if (row >= nrows) return;
    int lane = threadIdx.x & 31;
    float* hr = Hio + (size_t)row * d;

    float s = 0.0f, ss = 0.0f;
    for (int f = lane; f < d; f += 32) { float v = hr[f]; s += v; ss += v * v; }
    for (int off = 16; off; off >>= 1) {
        s  += __shfl_xor(s,  off, 32);
        ss += __shfl_xor(ss, off, 32);
    }
    float mean = s / (float)d;
    float var  = ss / (float)d - mean * mean;
    float rstd = rsqrtf(var + 1e-5f);

    for (int f = lane; f < d; f += 32) {
        float v = (hr[f] - mean) * rstd * g[f] + be[f];
        if (act) v = v / (1.0f + expf(-v));          // SiLU
        hr[f] = v;
    }
}

__global__ void k_pool_count(const int* __restrict__ batch, float* __restrict__ cnt, int nrows) {
    int i = blockIdx.x * blockDim.x + threadIdx.x;
    if (i < nrows) atomicAdd(&cnt[batch[i]], 1.0f);
}

__global__ void k_pool_accum(const float* __restrict__ H, const int* __restrict__ batch,
                             float* __restrict__ out, int nrows, int d)
{
    const int dq = d >> 2;
    long long idx = (long long)blockIdx.x * blockDim.x + threadIdx.x;
    if (idx >= (long long)nrows * dq) return;
    int row = (int)(idx / dq);
    int f4  = (int)(idx % dq) << 2;
    int gidx = batch[row];
    v4f h = *(const v4f*)&H[(size_t)row * d + f4];
    float* o = &out[(size_t)gidx * d + f4];
    atomicAdd(o + 0, h.x);
    atomicAdd(o + 1, h.y);
    atomicAdd(o + 2, h.z);
    atomicAdd(o + 3, h.w);
}

__global__ void k_pool_div(float* __restrict__ out, const float* __restrict__ cnt,
                           int ng, int d)
{
    int idx = blockIdx.x * blockDim.x + threadIdx.x;
    if (idx < ng * d) out[idx] /= fmaxf(cnt[idx / d], 1.0f);
}

// ---------------------------------------------------------------------------
// host side
// ---------------------------------------------------------------------------
static void run_gemm(const float* X, const float* W, float* H,
                     int nrows, int K, int ncols, hipStream_t s)
{
    dim3 grid((nrows + 127) / 128, ncols / 64);
    gemm_wmma_f32<<<grid, 256, 0, s>>>(X, W, H, nrows, K, ncols);
}

static void run_block(const float* X, int din, int dh,
                      const float* W, const float* b,
                      const float* Rw, const float* Rb,
                      const float* g, const float* be,
                      float* Hbuf, float* Obuf,
                      const int* ei, const float* dinv, const float* coef,
                      int act, hipStream_t s)
{
    run_gemm(X, W,  Hbuf, NNODES, din, dh, s);   // H = X @ W
    run_gemm(X, Rw, Obuf, NNODES, din, dh, s);   // O = X @ Rw
    long long nd = (long long)NNODES * dh;
    k_combine<<<(unsigned)((nd + 255) / 256), 256, 0, s>>>(Obuf, Hbuf, dinv, b, Rb, NNODES, dh);
    long long ed = (long long)NEDGES * (dh >> 2);
    k_scatter<<<(unsigned)((ed + 255) / 256), 256, 0, s>>>(Hbuf, ei, coef, Obuf, NEDGES, dh);
    k_layernorm<<<(NNODES + 7) / 8, 256, 0, s>>>(Obuf, g, be, NNODES, dh, act);
}

extern "C" void kernel_launch(void* const* d_in, const int* in_sizes, int n_in,
                              void* d_out, int out_size, void* d_ws, size_t ws_size,
                              hipStream_t stream)
{
    // inputs in setup_inputs() order
    const float* x     = (const float*)d_in[0];
    const int*   ei    = (const int*)  d_in[1];
    const int*   batch = (const int*)  d_in[2];
    const float* W1 = (const float*)d_in[3];  const float* b1 = (const float*)d_in[4];
    const float* W2 = (const float*)d_in[5];  const float* b2 = (const float*)d_in[6];
    const float* W3 = (const float*)d_in[7];  const float* b3 = (const float*)d_in[8];
    const float* R1w = (const float*)d_in[9];  const float* R1b = (const float*)d_in[10];
    const float* R2w = (const float*)d_in[11]; const float* R2b = (const float*)d_in[12];
    const float* R3w = (const float*)d_in[13]; const float* R3b = (const float*)d_in[14];
    const float* g1 = (const float*)d_in[15]; const float* be1 = (const float*)d_in[16];
    const float* g2 = (const float*)d_in[17]; const float* be2 = (const float*)d_in[18];
    const float* g3 = (const float*)d_in[19]; const float* be3 = (const float*)d_in[20];
    float* out = (float*)d_out;

    // workspace layout (floats)
    float* ws   = (float*)d_ws;
    float* deg  = ws;                               // N   (becomes dinv)
    float* cnt  = deg + NNODES;                     // G
    float* coef = cnt + NGRAPH;                     // E
    float* Hbuf = coef + NEDGES;                    // N*256
    float* B0   = Hbuf + (size_t)NNODES * DH;       // N*256
    float* B1   = B0   + (size_t)NNODES * DH;       // N*256

    // --- degrees & normalization coefficients ---
    k_zero<<<(NNODES + 255) / 256, 256, 0, stream>>>(deg, NNODES);
    k_count_deg<<<(NEDGES + 255) / 256, 256, 0, stream>>>(ei, deg, NEDGES);
    k_deg_to_dinv<<<(NNODES + 255) / 256, 256, 0, stream>>>(deg, NNODES);
    k_edge_coef<<<(NEDGES + 255) / 256, 256, 0, stream>>>(ei, deg, coef, NEDGES);

    // --- block 1: x(128) -> B0(256), LN + SiLU ---
    run_block(x, DIN, DH, W1, b1, R1w, R1b, g1, be1, Hbuf, B0, ei, deg, coef, 1, stream);
    // --- block 2: B0(256) -> B1(256), LN + SiLU ---
    run_block(B0, DH, DH, W2, b2, R2w, R2b, g2, be2, Hbuf, B1, ei, deg, coef, 1, stream);
    // --- block 3: B1(256) -> B0(128), LN only ---
    run_block(B1, DH, DOUT, W3, b3, R3w, R3b, g3, be3, Hbuf, B0, ei, deg, coef, 0, stream);

    // --- global mean pool ---
    k_zero<<<(NGRAPH * DOUT + 255) / 256, 256, 0, stream>>>(out, NGRAPH * DOUT);
    k_zero<<<1, 256, 0, stream>>>(cnt, NGRAPH);
    k_pool_count<<<(NNODES + 255) / 256, 256, 0, stream>>>(batch, cnt, NNODES);
    long long ndo = (long long)NNODES * (DOUT >> 2);
    k_pool_accum<<<(unsigned)((ndo + 255) / 256), 256, 0, stream>>>(B0, batch, out, NNODES, DOUT);
    k_pool_div<<<(NGRAPH * DOUT + 255) / 256, 256, 0, stream>>>(out, cnt, NGRAPH, DOUT);
}